// PoseGNN_16896401342873
// MI455X (gfx1250) — compile-verified
//
#include <hip/hip_runtime.h>

typedef __attribute__((ext_vector_type(2))) float v2f;
typedef __attribute__((ext_vector_type(8))) float v8f;

// Relaxed agent-scope fp32 atomic add -> global_atomic_add_f32 on gfx1250
#define ATOM_ADD(p, v) __hip_atomic_fetch_add((p), (v), __ATOMIC_RELAXED, __HIP_MEMORY_SCOPE_AGENT)
#define ATOM_ADD_I32(p, v) __hip_atomic_fetch_add((p), (v), __ATOMIC_RELAXED, __HIP_MEMORY_SCOPE_AGENT)

// gfx1250 async global->LDS copy (16B per active lane), tracked by ASYNCcnt.
__device__ __forceinline__ void async_g2lds_b128(const float* gsrc, unsigned lds_byte_off) {
  asm volatile("global_load_async_to_lds_b128 %0, %1, off"
               :: "v"(lds_byte_off), "v"(gsrc)
               : "memory");
}
__device__ __forceinline__ void wait_asynccnt0() {
  asm volatile("s_wait_asynccnt 0x0" ::: "memory");
}

// ---------------------------------------------------------------- utilities
__global__ void fill_f32_kernel(float* __restrict__ p, long long n, float val) {
  long long t = (long long)blockIdx.x * blockDim.x + threadIdx.x;
  if (t < n) p[t] = val;
}
__global__ void fill_i32_kernel(int* __restrict__ p, int n, int val) {
  int t = blockIdx.x * blockDim.x + threadIdx.x;
  if (t < n) p[t] = val;
}

__global__ void deg_count_kernel(const int* __restrict__ cols,
                                 int* __restrict__ deg, int E) {
  int t = blockIdx.x * blockDim.x + threadIdx.x;
  if (t < E) ATOM_ADD_I32(&deg[cols[t]], 1);
}

// dinv = rsqrt(deg + 1)   (+1 = self loop)
__global__ void dinv_kernel(const int* __restrict__ deg, float* __restrict__ dinv, int n) {
  int t = blockIdx.x * blockDim.x + threadIdx.x;
  if (t < n) dinv[t] = rsqrtf((float)deg[t] + 1.0f);
}

// --------------------------- hierarchical exclusive scan (256 elems/block)
__global__ void scan_block_kernel(const int* __restrict__ in, int* __restrict__ out,
                                  int* __restrict__ bsums, int n) {
  __shared__ int tmp[256];
  const int tid = threadIdx.x;
  const int t = blockIdx.x * 256 + tid;
  int v = (t < n) ? in[t] : 0;
  tmp[tid] = v;
  __syncthreads();
  for (int ofs = 1; ofs < 256; ofs <<= 1) {
    int x = (tid >= ofs) ? tmp[tid - ofs] : 0;
    __syncthreads();
    tmp[tid] += x;
    __syncthreads();
  }
  int incl = tmp[tid];
  if (t < n) out[t] = incl - v;                 // exclusive
  if (tid == 255) bsums[blockIdx.x] = incl;
}

__global__ void scan_aux_kernel(int* __restrict__ bsums, int nb) {  // 1 block, 1024 thr
  __shared__ int tmp[1024];
  const int tid = threadIdx.x;
  int v = (tid < nb) ? bsums[tid] : 0;
  tmp[tid] = v;
  __syncthreads();
  for (int ofs = 1; ofs < 1024; ofs <<= 1) {
    int x = (tid >= ofs) ? tmp[tid - ofs] : 0;
    __syncthreads();
    tmp[tid] += x;
    __syncthreads();
  }
  if (tid < nb) bsums[tid] = tmp[tid] - v;      // exclusive
}

__global__ void scan_add_kernel(int* __restrict__ out, const int* __restrict__ bsums, int n) {
  int t = blockIdx.x * 256 + threadIdx.x;
  if (t < n) out[t] += bsums[blockIdx.x];
}

// CSR fill: esrc[cursor[col[e]]++] = row[e]
__global__ void fill_csr_kernel(const int* __restrict__ rows, const int* __restrict__ cols,
                                int* __restrict__ cursor, int* __restrict__ esrc, int E) {
  int e = blockIdx.x * blockDim.x + threadIdx.x;
  if (e < E) {
    int p = ATOM_ADD_I32(&cursor[cols[e]], 1);
    esrc[p] = rows[e];
  }
}

// ------------------------------------------------------- WMMA fp32 GEMM
// out[m, n] = dinv[m] * sum_k A[m,k] * B[k,n]
// Block = (NOUT/16) waves; wave w owns a 16-col slice of B in registers
// (64 VGPRs for K=128). Block iterates MT m-tiles; 16xK A tiles are staged
// into a double-buffered LDS tile via async global->LDS DMA so the next
// tile's DMA overlaps the current tile's DS+WMMA stream.
template <int K>
__device__ __forceinline__ void stage_tile(const float* __restrict__ A, int M, int m0,
                                           float* dst, int tid, int nthr) {
  constexpr int LDA = K + 4;
  for (int idx = tid; idx < 16 * (K / 4); idx += nthr) {
    const int row = idx / (K / 4);
    const int c4  = (idx % (K / 4)) * 4;
    int grow = m0 + row;
    if (grow >= M) grow = M - 1;                // tail-safe
    async_g2lds_b128(A + (size_t)grow * K + c4,
                     (unsigned)(uintptr_t)&dst[row * LDA + c4]);
  }
}

template <int K>
__launch_bounds__(256)
__global__ void gemm_wmma_scale(const float* __restrict__ A,
                                const float* __restrict__ B,
                                const float* __restrict__ dinv,
                                float* __restrict__ out,
                                int M, int NOUT, int MT) {
  constexpr int LDA = K + 4;                    // (K+4) % 64 == 4 -> no bank conflicts
  __shared__ float tileA[2][16 * LDA];

  const int lane = threadIdx.x & 31;
  const int wave = threadIdx.x >> 5;
  const int n0   = wave * 16;
  const int r    = lane & 15;                   // row (A) / col (B,C) in tile
  const int half = lane >> 4;                   // which K-pair this lane holds

  const int totTiles = (M + 15) >> 4;
  const int baseTile = blockIdx.x * MT;
  const int cnt = min(MT, totTiles - baseTile);
  if (cnt <= 0) return;

  // ---- preload this wave's B slice into registers (reused for all m-tiles)
  v2f bfrag[K / 4];
#pragma unroll
  for (int k = 0; k < K; k += 4) {
    const int ka = k + half * 2;
    bfrag[k >> 2].x = B[(size_t)ka * NOUT + n0 + r];
    bfrag[k >> 2].y = B[(size_t)(ka + 1) * NOUT + n0 + r];
  }

  // ---- prologue: stage first tile
  stage_tile<K>(A, M, baseTile * 16, tileA[0], threadIdx.x, blockDim.x);
  wait_asynccnt0();
  __syncthreads();

  for (int mt = 0; mt < cnt; ++mt) {
    const int m0 = (baseTile + mt) * 16;
    // overlap: DMA next tile into the other buffer while computing this one
    if (mt + 1 < cnt)
      stage_tile<K>(A, M, m0 + 16, tileA[(mt + 1) & 1], threadIdx.x, blockDim.x);

    const float* tA = tileA[mt & 1];
    v8f acc = {};
#pragma unroll
    for (int k = 0; k < K; k += 4) {
      v2f a = *(const v2f*)&tA[r * LDA + k + half * 2];  // ds_load_b64
      acc = __builtin_amdgcn_wmma_f32_16x16x4_f32(
          false, a, false, bfrag[k >> 2], (short)0, acc, false, false);
    }

#pragma unroll
    for (int vr = 0; vr < 8; ++vr) {            // C/D VGPR vr = rows vr / vr+8
      const int row = m0 + vr + half * 8;
      if (row < M)
        out[(size_t)row * NOUT + n0 + r] = dinv[row] * acc[vr];
    }
    wait_asynccnt0();                           // own DMAs for next buffer done
    __syncthreads();                            // everyone done reading/writing
  }
}

// -------------------------- CSR gather, layer-1 epilogue: relu(s*sum + b)
// thread t handles (node i = t>>lgc, 4-float chunk f); self-loop included.
__global__ void gather_relu_kernel(const float* __restrict__ xs,
                                   const int* __restrict__ rptr,
                                   const int* __restrict__ deg,
                                   const int* __restrict__ esrc,
                                   const float* __restrict__ dinv,
                                   const float* __restrict__ bias,
                                   float* __restrict__ out,
                                   long long total, int F, int lgc) {
  long long t = (long long)blockIdx.x * blockDim.x + threadIdx.x;
  if (t >= total) return;
  const int i = (int)(t >> lgc);
  const int f = ((int)t & ((1 << lgc) - 1)) << 2;
  float4 s = *(const float4*)(xs + (size_t)i * F + f);   // self loop
  const int b = rptr[i];
  const int d = deg[i];
  for (int j = 0; j < d; ++j) {
    const float4 v = *(const float4*)(xs + (size_t)esrc[b + j] * F + f);
    s.x += v.x; s.y += v.y; s.z += v.z; s.w += v.w;
  }
  const float sc = dinv[i];
  float4 o;
  o.x = fmaxf(fmaf(sc, s.x, bias[f + 0]), 0.0f);
  o.y = fmaxf(fmaf(sc, s.y, bias[f + 1]), 0.0f);
  o.z = fmaxf(fmaf(sc, s.z, bias[f + 2]), 0.0f);
  o.w = fmaxf(fmaf(sc, s.w, bias[f + 3]), 0.0f);
  *(float4*)(out + (size_t)i * F + f) = o;
}

// ---------------- CSR gather, layer-2 epilogue fused with mean-pool sums
__global__ void gather_pool_kernel(const float* __restrict__ hs,
                                   const int* __restrict__ rptr,
                                   const int* __restrict__ deg,
                                   const int* __restrict__ esrc,
                                   const float* __restrict__ dinv,
                                   const float* __restrict__ bias,
                                   const int* __restrict__ batch,
                                   float* __restrict__ sums, float* __restrict__ cnts,
                                   long long total, int F, int lgc) {
  long long t = (long long)blockIdx.x * blockDim.x + threadIdx.x;
  if (t >= total) return;
  const int i = (int)(t >> lgc);
  const int f = ((int)t & ((1 << lgc) - 1)) << 2;
  float4 s = *(const float4*)(hs + (size_t)i * F + f);   // self loop
  const int b = rptr[i];
  const int d = deg[i];
  for (int j = 0; j < d; ++j) {
    const float4 v = *(const float4*)(hs + (size_t)esrc[b + j] * F + f);
    s.x += v.x; s.y += v.y; s.z += v.z; s.w += v.w;
  }
  const float sc = dinv[i];
  const int g = batch[i];
  float* q = sums + (size_t)g * F + f;
  ATOM_ADD(q + 0, fmaf(sc, s.x, bias[f + 0]));
  ATOM_ADD(q + 1, fmaf(sc, s.y, bias[f + 1]));
  ATOM_ADD(q + 2, fmaf(sc, s.z, bias[f + 2]));
  ATOM_ADD(q + 3, fmaf(sc, s.w, bias[f + 3]));
  if (f == 0) ATOM_ADD(&cnts[g], 1.0f);
}

__global__ void mean_kernel(const float* __restrict__ sums,
                            const float* __restrict__ cnts,
                            float* __restrict__ out, int total, int F) {
  int t = blockIdx.x * blockDim.x + threadIdx.x;
  if (t < total) out[t] = sums[t] / fmaxf(cnts[t / F], 1.0f);
}

// ---------------------------------------------------------------- launcher
static inline int cdivll(long long a, long long b) { return (int)((a + b - 1) / b); }

extern "C" void kernel_launch(void* const* d_in, const int* in_sizes, int n_in,
                              void* d_out, int out_size, void* d_ws, size_t ws_size,
                              hipStream_t stream) {
  const float* x     = (const float*)d_in[0];
  const int*   ei    = (const int*)d_in[1];   // [2, E]
  const int*   batch = (const int*)d_in[2];
  const float* W1    = (const float*)d_in[3];
  const float* b1    = (const float*)d_in[4];
  const float* W2    = (const float*)d_in[5];
  const float* b2    = (const float*)d_in[6];

  const int Nn  = in_sizes[2];
  const int E   = in_sizes[1] / 2;
  const int HID = in_sizes[4];            // 128
  const int OUT = in_sizes[6];            // 64
  const int G   = out_size / OUT;         // 256
  const int MT  = 8;                      // m-tiles per GEMM block

  const int* rows = ei;                   // sources
  const int* cols = ei + E;               // targets

  // ---------------- workspace layout: ints first, then 256B-aligned floats
  const int nScanBlocks = cdivll(Nn, 256);
  int* deg_i  = (int*)d_ws;               // N
  int* rptr   = deg_i + Nn;               // N
  int* cursor = rptr + Nn;                // N
  int* esrc   = cursor + Nn;              // E
  int* bsums  = esrc + E;                 // <= 1024
  size_t off = ((size_t)(bsums + 1024 - (int*)d_ws) + 63) & ~(size_t)63;
  float* ws   = (float*)d_ws;
  float* dinv = ws + off; off += (size_t)Nn;          off = (off + 63) & ~(size_t)63;
  float* xs   = ws + off; off += (size_t)Nn * HID;    off = (off + 63) & ~(size_t)63;
  float* h    = ws + off; off += (size_t)Nn * HID;    off = (off + 63) & ~(size_t)63;
  float* hs   = ws + off; off += (size_t)Nn * OUT;    off = (off + 63) & ~(size_t)63;
  float* sums = ws + off; off += (size_t)G * OUT;
  float* cnts = ws + off; off += (size_t)G;
  (void)ws_size; (void)n_in;

  const int B = 256;

  // ---------------- degrees + CSR of incoming edges (built once, reused)
  fill_i32_kernel<<<cdivll(Nn, B), B, 0, stream>>>(deg_i, Nn, 0);
  fill_f32_kernel<<<cdivll((long long)G * OUT + G, B), B, 0, stream>>>(
      sums, (long long)G * OUT + G, 0.0f);
  deg_count_kernel<<<cdivll(E, B), B, 0, stream>>>(cols, deg_i, E);
  dinv_kernel<<<cdivll(Nn, B), B, 0, stream>>>(deg_i, dinv, Nn);
  scan_block_kernel<<<nScanBlocks, 256, 0, stream>>>(deg_i, rptr, bsums, Nn);
  scan_aux_kernel<<<1, 1024, 0, stream>>>(bsums, nScanBlocks);
  scan_add_kernel<<<nScanBlocks, 256, 0, stream>>>(rptr, bsums, Nn);
  hipMemcpyAsync(cursor, rptr, (size_t)Nn * sizeof(int), hipMemcpyDeviceToDevice, stream);
  fill_csr_kernel<<<cdivll(E, B), B, 0, stream>>>(rows, cols, cursor, esrc, E);

  // ---------------- layer 1: xs = dinv .* (x @ W1); h = relu(dinv.*gather + b1)
  gemm_wmma_scale<128><<<cdivll(Nn, 16 * MT), (HID / 16) * 32, 0, stream>>>(
      x, W1, dinv, xs, Nn, HID, MT);
  {
    long long total = (long long)Nn << 5;  // F=128 -> 32 float4 chunks
    gather_relu_kernel<<<cdivll(total, B), B, 0, stream>>>(
        xs, rptr, deg_i, esrc, dinv, b1, h, total, HID, 5);
  }

  // ---------------- layer 2: hs = dinv .* (h @ W2); gather + bias + pool
  gemm_wmma_scale<128><<<cdivll(Nn, 16 * MT), (OUT / 16) * 32, 0, stream>>>(
      h, W2, dinv, hs, Nn, OUT, MT);
  {
    long long total = (long long)Nn << 4;  // F=64 -> 16 float4 chunks
    gather_pool_kernel<<<cdivll(total, B), B, 0, stream>>>(
        hs, rptr, deg_i, esrc, dinv, b2, batch, sums, cnts, total, OUT, 4);
  }
  mean_kernel<<<cdivll(G * OUT, B), B, 0, stream>>>(sums, cnts, (float*)d_out, G * OUT, OUT);
}